// KANLayer_72739566125259
// MI455X (gfx1250) — compile-verified
//
#include <hip/hip_runtime.h>
#include <hip/hip_bf16.h>

// ---------------------------------------------------------------------------
// KAN layer on MI455X (gfx1250, wave32).
//   out[b,o] = sum_k A_ext[b,k] * W_ext[o,k],  K = 896 = 128*6 + 128
//   A_ext[b, i*6+m]  = B_m(x[b,i])        (cubic B-spline basis, 6 funcs)
//   A_ext[b, 768+i]  = swish(x[b,i])
//   W_ext[o, i*6+m]  = c_spl[o*128+i] * c_basis[o*128+i, m] / 128
//   W_ext[o, 768+i]  = c_res[o*128+i] / 128
// GEMM done with exact-fp32 V_WMMA_F32_16X16X4_F32 (problem is tiny /
// L2-resident, so precision > low-precision throughput).
// ---------------------------------------------------------------------------

#define N_IN   128
#define N_OUT  256
#define BATCH  512
#define KDIM   896          // 768 spline cols + 128 residual cols
#define NKNOT  10           // G + 2K + 1

typedef __attribute__((ext_vector_type(2))) float v2f;
typedef __attribute__((ext_vector_type(8))) float v8f;

// ---------------- Kernel 1: activation matrix A_ext (512 x 896) -------------
__global__ void kan_build_A(const float* __restrict__ x,
                            const float* __restrict__ grid,   // row 0: 10 knots
                            float* __restrict__ A)
{
    int idx = blockIdx.x * blockDim.x + threadIdx.x;   // 512*128 threads
    if (idx >= BATCH * N_IN) return;
    int b = idx / N_IN;
    int i = idx % N_IN;

    float t = x[b * N_IN + i];

    // knots (uniform row, all grid rows identical -> read row 0)
    float g[NKNOT];
#pragma unroll
    for (int p = 0; p < NKNOT; ++p) g[p] = grid[p];

    // Cox–de Boor, identical to reference: deg-0 strict indicators then k=1..3
    float bas[NKNOT - 1];
#pragma unroll
    for (int p = 0; p < NKNOT - 1; ++p)
        bas[p] = (t >= g[p] && t < g[p + 1]) ? 1.0f : 0.0f;

#pragma unroll
    for (int kk = 1; kk <= 3; ++kk) {
#pragma unroll
        for (int p = 0; p <= 8 - kk; ++p) {
            float left  = (t - g[p])          / (g[p + kk]     - g[p])     * bas[p];
            float right = (g[p + kk + 1] - t) / (g[p + kk + 1] - g[p + 1]) * bas[p + 1];
            bas[p] = left + right;
        }
    }

    float sig   = 1.0f / (1.0f + __expf(-t));
    float swish = t * sig;

    float* Arow = A + (size_t)b * KDIM;
#pragma unroll
    for (int m = 0; m < 6; ++m) Arow[i * 6 + m] = bas[m];
    Arow[768 + i] = swish;
}

// ---------------- Kernel 2: weight matrix W_ext (256 x 896) -----------------
__global__ void kan_build_W(const float* __restrict__ c_basis,  // (32768, 6)
                            const float* __restrict__ c_spl,    // (32768,)
                            const float* __restrict__ c_res,    // (32768,)
                            float* __restrict__ W)
{
    int idx = blockIdx.x * blockDim.x + threadIdx.x;   // 256*896 threads
    if (idx >= N_OUT * KDIM) return;
    int o  = idx / KDIM;
    int kk = idx % KDIM;

    float val;
    if (kk < 768) {
        int i = kk / 6;
        // c_basis viewed as (256, 768) row-major is exactly W's spline block
        val = c_spl[o * N_IN + i] * c_basis[(size_t)o * 768 + kk];
    } else {
        val = c_res[o * N_IN + (kk - 768)];
    }
    W[(size_t)o * KDIM + kk] = val * (1.0f / (float)N_IN);
}

// ---------------- Kernel 3: WMMA fp32 GEMM  out = A_ext * W_ext^T -----------
// 512 output tiles of 16x16 (32 M-tiles x 16 N-tiles); 1 tile per wave32;
// 8 waves / block -> 64 blocks. K loop: 224 steps of V_WMMA_F32_16X16X4_F32.
__global__ void __launch_bounds__(256)
kan_wmma_gemm(const float* __restrict__ A,   // (512, 896)
              const float* __restrict__ W,   // (256, 896)
              float* __restrict__ out)       // (512, 256)
{
    const int wave = threadIdx.x >> 5;
    const int lane = threadIdx.x & 31;
    const int half = lane >> 4;        // 0: lanes 0-15, 1: lanes 16-31
    const int r    = lane & 15;

    const int tile  = blockIdx.x * 8 + wave;   // 0..511
    const int mbase = (tile & 31) * 16;        // batch-tile
    const int nbase = (tile >> 5) * 16;        // out-feature tile

    const float* Arow = A + (size_t)(mbase + r) * KDIM;
    const float* Wrow = W + (size_t)(nbase + r) * KDIM;

    v8f acc = {};

    // A fragment layout (16x4 f32): lanes 0-15 hold K = k0+0,k0+1 ; lanes
    // 16-31 hold K = k0+2,k0+3 (mirrored for B 4x16). Both are contiguous
    // float2 loads per lane.
#pragma unroll 4
    for (int k0 = 0; k0 < KDIM; k0 += 4) {
        v2f a = *(const v2f*)(Arow + k0 + 2 * half);
        v2f b = *(const v2f*)(Wrow + k0 + 2 * half);
        acc = __builtin_amdgcn_wmma_f32_16x16x4_f32(
                  /*neg_a=*/false, a, /*neg_b=*/false, b,
                  /*c_mod=*/(short)0, acc,
                  /*reuse_a=*/false, /*reuse_b=*/false);
    }

    // C/D layout: VGPR v -> row mbase + v + 8*half, col nbase + r
    const int col = nbase + r;
#pragma unroll
    for (int v = 0; v < 8; ++v) {
        out[(size_t)(mbase + v + 8 * half) * N_OUT + col] = acc[v];
    }
}

// ---------------------------------------------------------------------------
extern "C" void kernel_launch(void* const* d_in, const int* in_sizes, int n_in,
                              void* d_out, int out_size, void* d_ws, size_t ws_size,
                              hipStream_t stream)
{
    const float* x       = (const float*)d_in[0];   // (512, 128)
    const float* grid    = (const float*)d_in[1];   // (32768, 10), rows identical
    const float* c_basis = (const float*)d_in[2];   // (32768, 6)
    const float* c_spl   = (const float*)d_in[3];   // (32768,)
    const float* c_res   = (const float*)d_in[4];   // (32768,)
    float*       out     = (float*)d_out;           // (512, 256)

    float* A = (float*)d_ws;                               // 512*896 f32
    float* W = A + (size_t)BATCH * KDIM;                   // 256*896 f32
    // total ws use: (512+256)*896*4 = 2.75 MB

    kan_build_A<<<(BATCH * N_IN + 255) / 256, 256, 0, stream>>>(x, grid, A);
    kan_build_W<<<(N_OUT * KDIM + 255) / 256, 256, 0, stream>>>(c_basis, c_spl, c_res, W);
    kan_wmma_gemm<<<64, 256, 0, stream>>>(A, W, out);
}